// InvariantAttention_87462714016124
// MI455X (gfx1250) — compile-verified
//
#include <hip/hip_runtime.h>

#define BB 4
#define TT 512
#define CC 768
#define HH 24
#define DD 32
#define SS 1024

typedef __attribute__((ext_vector_type(2))) float v2f;
typedef __attribute__((ext_vector_type(8))) float v8f;

__device__ __forceinline__ v8f wmma_f32(v2f a, v2f b, v8f c) {
  // D = A(16x4 f32) * B(4x16 f32) + C(16x16 f32)
  return __builtin_amdgcn_wmma_f32_16x16x4_f32(
      /*neg_a=*/false, a, /*neg_b=*/false, b,
      /*c_mod=*/(short)0, c, /*reuse_a=*/false, /*reuse_b=*/false);
}

// ---------------------------------------------------------------------------
// Kernel 1: gather K/V to S=2T, RoPE(Q scaled), RoPE(K), write packed layouts.
//   qh : [B*H][T][D]                     row-major (A-fragment friendly)
//   kT2: [B*H][D/2][S][2]  (d-pair packed K^T -> B-fragment = one float2/lane)
//   vh2: [B*H][S/2][D][2]  (s-pair packed V   -> B-fragment = one float2/lane)
// One thread handles the RoPE pair (d, d+16); idx spans B*S*H*16.
// ---------------------------------------------------------------------------
__global__ void prep_kernel(const float* __restrict__ q,
                            const float* __restrict__ k,
                            const float* __restrict__ v,
                            const int* __restrict__ outcell,
                            float* __restrict__ qh,
                            float* __restrict__ kT2,
                            float* __restrict__ vh2) {
  int idx = blockIdx.x * blockDim.x + threadIdx.x;
  const int total = BB * SS * HH * 16;
  if (idx >= total) return;
  int d = idx & 15;
  int h = (idx >> 4) % HH;
  int s = (idx / (16 * HH)) % SS;
  int b = idx / (16 * HH * SS);

  int src_t = (s < TT) ? s : outcell[b * TT + (s - TT)];
  const float* krow = k + ((size_t)(b * TT + src_t)) * CC + h * DD;
  const float* vrow = v + ((size_t)(b * TT + src_t)) * CC + h * DD;
  float k1 = krow[d], k2 = krow[d + 16];
  float v1 = vrow[d], v2 = vrow[d + 16];

  // RoPE: inv_freq = 10000^(-2d/32); same angle for d and d+16
  float inv_freq = __powf(10000.f, -(float)(2 * d) / (float)DD);
  float ang = (float)s * inv_freq;
  float sn, cs;
  __sincosf(ang, &sn, &cs);
  float kr1 = k1 * cs - k2 * sn;
  float kr2 = k2 * cs + k1 * sn;

  int bh = b * HH + h;
  kT2[((size_t)(bh * 16 + (d >> 1)) * SS + s) * 2 + (d & 1)] = kr1;
  kT2[((size_t)(bh * 16 + ((d + 16) >> 1)) * SS + s) * 2 + (d & 1)] = kr2;
  vh2[((size_t)(bh * (SS / 2) + (s >> 1)) * DD + d) * 2 + (s & 1)] = v1;
  vh2[((size_t)(bh * (SS / 2) + (s >> 1)) * DD + d + 16) * 2 + (s & 1)] = v2;

  if (s < TT) {
    const float scal = 0.17677669529663688f;  // sqrt(D)/D = 1/sqrt(32)
    const float* qrow = q + ((size_t)(b * TT + s)) * CC + h * DD;
    float q1 = qrow[d] * scal, q2 = qrow[d + 16] * scal;
    float* qd = qh + ((size_t)(bh * TT + s)) * DD;
    qd[d] = q1 * cs - q2 * sn;
    qd[d + 16] = q2 * cs + q1 * sn;
  }
}

// ---------------------------------------------------------------------------
// W_out pre-transpose + k-pair pack: Wt2[(k/2)*C + n][2] = W_out[n][k]
// ---------------------------------------------------------------------------
__global__ void wt_kernel(const float* __restrict__ W, float* __restrict__ Wt2) {
  int idx = blockIdx.x * blockDim.x + threadIdx.x;
  if (idx >= CC * CC) return;
  int kq = idx / CC;
  int n = idx % CC;
  Wt2[((size_t)(kq >> 1) * CC + n) * 2 + (kq & 1)] = W[(size_t)n * CC + kq];
}

// ---------------------------------------------------------------------------
// Kernel 2: flash attention, one wave32 per (b, h, 16-row tile of t).
// Streams 64 key columns per iteration (4 WMMA score tiles) so the
// shuffle-tree softmax reductions are amortized 4x. Bias goes straight into
// the WMMA C accumulators; P relayout to A-fragments goes through a
// bank-conflict-padded LDS buffer (row stride 66 dwords).
// ---------------------------------------------------------------------------
__global__ __launch_bounds__(32) void attn_kernel(
    const float* __restrict__ qh, const float* __restrict__ kT2,
    const float* __restrict__ vh2, const float* __restrict__ bias,
    const float* __restrict__ law, const unsigned char* __restrict__ kpm,
    const unsigned char* __restrict__ epm, float* __restrict__ attn) {
  __shared__ float pbuf[16 * 66];  // 16 rows x 64 cols, stride 66 (padding)
  const int lane = threadIdx.x;
  const int ml = lane & 15;
  const int half = lane >> 4;

  int blk = blockIdx.x;
  int t0 = (blk % (TT / 16)) * 16;
  int h = (blk / (TT / 16)) % HH;
  int b = blk / ((TT / 16) * HH);
  int bh = b * HH + h;

  // Preload Q A-fragments (16x32 tile -> 8 K-chunks of 4)
  v2f qa[8];
  const float* qbase = qh + ((size_t)(bh * TT + t0 + ml)) * DD;
#pragma unroll
  for (int kc = 0; kc < 8; ++kc)
    qa[kc] = *(const v2f*)(qbase + kc * 4 + half * 2);

  const v8f vzero = {0.f, 0.f, 0.f, 0.f, 0.f, 0.f, 0.f, 0.f};
  v8f o0 = vzero, o1 = vzero;
  float mrow[8], lrow[8];
#pragma unroll
  for (int r = 0; r < 8; ++r) { mrow[r] = -1e30f; lrow[r] = 0.f; }

  const float* biasBase = bias + ((size_t)(bh * TT + t0)) * SS;
  const float* lawBase = law + ((size_t)(b * TT + t0)) * SS;

  for (int s0 = 0; s0 < SS; s0 += 64) {
    // 4 score tiles; C accumulators initialized with the bias tiles
    v8f acc[4];
#pragma unroll
    for (int st = 0; st < 4; ++st)
#pragma unroll
      for (int r = 0; r < 8; ++r)
        acc[st][r] = biasBase[(size_t)(r + half * 8) * SS + s0 + st * 16 + ml];

    // scores = Q * K^T + bias  (A fragment reused across the 4 tiles)
#pragma unroll
    for (int kc = 0; kc < 8; ++kc) {
      const float* kb =
          kT2 + ((size_t)(bh * 16 + kc * 2 + half) * SS + s0 + ml) * 2;
#pragma unroll
      for (int st = 0; st < 4; ++st) {
        v2f bf = *(const v2f*)(kb + st * 32);  // +16 columns per tile
        acc[st] = wmma_f32(qa[kc], bf, acc[st]);
      }
    }

    // masks (column padding mask + local-attention cutoff); keep law in regs
    float lawv[4][8];
#pragma unroll
    for (int st = 0; st < 4; ++st) {
      int col = s0 + st * 16 + ml;
      bool masked = (col < TT) ? (kpm[b * TT + col] != 0)
                               : (epm[b * TT + col - TT] != 0);
#pragma unroll
      for (int r = 0; r < 8; ++r) {
        float lw = lawBase[(size_t)(r + half * 8) * SS + col];
        lawv[st][r] = lw;
        if (masked || lw <= 1e-5f) acc[st][r] = -1e30f;
      }
    }

    // online softmax: one shuffle tree per row per 64 columns
#pragma unroll
    for (int r = 0; r < 8; ++r) {
      float mx =
          fmaxf(fmaxf(acc[0][r], acc[1][r]), fmaxf(acc[2][r], acc[3][r]));
#pragma unroll
      for (int off = 1; off < 16; off <<= 1)
        mx = fmaxf(mx, __shfl_xor(mx, off, 32));
      float mnew = fmaxf(mrow[r], mx);
      float psum = 0.f;
#pragma unroll
      for (int st = 0; st < 4; ++st) {
        float p = __expf(acc[st][r] - mnew);
        psum += p;
        pbuf[(r + half * 8) * 66 + st * 16 + ml] = p * lawv[st][r];
      }
#pragma unroll
      for (int off = 1; off < 16; off <<= 1)
        psum += __shfl_xor(psum, off, 32);
      float corr = __expf(mrow[r] - mnew);
      lrow[r] = lrow[r] * corr + psum;
      mrow[r] = mnew;
      o0[r] *= corr;
      o1[r] *= corr;
    }
    __syncthreads();

    // O += P * V  (K=64 -> 16 sub-chunks of 4; two N-tiles cover D=32)
#pragma unroll
    for (int kk = 0; kk < 16; ++kk) {
      v2f a = *(const v2f*)(&pbuf[ml * 66 + kk * 4 + half * 2]);
      int sp = ((s0 + kk * 4) >> 1) + half;  // s-pair index
      const float* vb = vh2 + ((size_t)(bh * (SS / 2) + sp) * DD) * 2;
      v2f b0 = *(const v2f*)(vb + (0 + ml) * 2);
      v2f b1 = *(const v2f*)(vb + (16 + ml) * 2);
      o0 = wmma_f32(a, b0, o0);
      o1 = wmma_f32(a, b1, o1);
    }
    __syncthreads();
  }

  // normalize by softmax denominator, scatter to [B][T][C]
#pragma unroll
  for (int r = 0; r < 8; ++r) {
    float inv = 1.f / lrow[r];
    size_t row = (size_t)(b * TT + t0 + r + half * 8) * CC + h * DD;
    attn[row + ml] = o0[r] * inv;
    attn[row + 16 + ml] = o1[r] * inv;
  }
}

// ---------------------------------------------------------------------------
// Kernel 3a: LayerNorm over C per row
// ---------------------------------------------------------------------------
__global__ __launch_bounds__(256) void ln_kernel(const float* __restrict__ attn,
                                                 const float* __restrict__ gamma,
                                                 const float* __restrict__ beta,
                                                 float* __restrict__ lnout) {
  __shared__ float rs[256], rs2[256];
  int row = blockIdx.x;
  int tid = threadIdx.x;
  const float* x = attn + (size_t)row * CC;
  float s = 0.f, s2 = 0.f;
  float vals[3];
#pragma unroll
  for (int i = 0; i < 3; ++i) {
    float v = x[tid + i * 256];
    vals[i] = v;
    s += v;
    s2 += v * v;
  }
  rs[tid] = s;
  rs2[tid] = s2;
  __syncthreads();
  for (int off = 128; off > 0; off >>= 1) {
    if (tid < off) {
      rs[tid] += rs[tid + off];
      rs2[tid] += rs2[tid + off];
    }
    __syncthreads();
  }
  float mu = rs[0] / CC;
  float var = rs2[0] / CC - mu * mu;
  float inv = rsqrtf(var + 1e-5f);
  float* y = lnout + (size_t)row * CC;
#pragma unroll
  for (int i = 0; i < 3; ++i) {
    int c = tid + i * 256;
    y[c] = (vals[i] - mu) * inv * gamma[c] + beta[c];
  }
}

// ---------------------------------------------------------------------------
// Kernel 3b: out = LN(attn) @ W_out^T  (M=2048, N=768, K=768), f32 WMMA.
// One wave computes a 16x64 strip (A fragment reused across 4 N-tiles).
// ---------------------------------------------------------------------------
__global__ __launch_bounds__(32) void gemm_kernel(const float* __restrict__ A,
                                                  const float* __restrict__ Wt2,
                                                  float* __restrict__ out) {
  const int lane = threadIdx.x;
  const int ml = lane & 15;
  const int half = lane >> 4;
  int m0 = blockIdx.x * 16;
  int n0 = blockIdx.y * 64;
  const v8f vzero = {0.f, 0.f, 0.f, 0.f, 0.f, 0.f, 0.f, 0.f};
  v8f acc[4];
#pragma unroll
  for (int nt = 0; nt < 4; ++nt) acc[nt] = vzero;

  const float* arow = A + (size_t)(m0 + ml) * CC;
  for (int kq = 0; kq < CC; kq += 4) {
    v2f a = *(const v2f*)(arow + kq + half * 2);
    const float* wb = Wt2 + ((size_t)((kq >> 1) + half) * CC) * 2;
#pragma unroll
    for (int nt = 0; nt < 4; ++nt) {
      v2f b = *(const v2f*)(wb + (n0 + nt * 16 + ml) * 2);
      acc[nt] = wmma_f32(a, b, acc[nt]);
    }
  }
#pragma unroll
  for (int nt = 0; nt < 4; ++nt)
#pragma unroll
    for (int r = 0; r < 8; ++r)
      out[(size_t)(m0 + r + half * 8) * CC + n0 + nt * 16 + ml] = acc[nt][r];
}

// ---------------------------------------------------------------------------
extern "C" void kernel_launch(void* const* d_in, const int* in_sizes, int n_in,
                              void* d_out, int out_size, void* d_ws,
                              size_t ws_size, hipStream_t stream) {
  const float* q = (const float*)d_in[0];
  const float* k = (const float*)d_in[1];
  const float* v = (const float*)d_in[2];
  const float* bias = (const float*)d_in[3];
  const float* law = (const float*)d_in[4];
  const float* W = (const float*)d_in[5];
  const float* gamma = (const float*)d_in[6];
  const float* beta = (const float*)d_in[7];
  const unsigned char* kpm = (const unsigned char*)d_in[8];
  const int* outcell = (const int*)d_in[9];
  const unsigned char* epm = (const unsigned char*)d_in[10];
  float* out = (float*)d_out;

  float* ws = (float*)d_ws;
  size_t off = 0;
  float* qh = ws + off;    off += (size_t)BB * HH * TT * DD;  // 6 MB
  float* kT2 = ws + off;   off += (size_t)BB * HH * DD * SS;  // 12 MB
  float* vh2 = ws + off;   off += (size_t)BB * HH * SS * DD;  // 12 MB
  float* attn = ws + off;  off += (size_t)BB * TT * CC;       // 6 MB
  float* lnout = ws + off; off += (size_t)BB * TT * CC;       // 6 MB
  float* Wt2 = ws + off;   off += (size_t)CC * CC;            // 2.25 MB

  prep_kernel<<<(BB * SS * HH * 16 + 255) / 256, 256, 0, stream>>>(
      q, k, v, outcell, qh, kT2, vh2);
  wt_kernel<<<(CC * CC + 255) / 256, 256, 0, stream>>>(W, Wt2);
  attn_kernel<<<BB * HH * (TT / 16), 32, 0, stream>>>(qh, kT2, vh2, bias, law,
                                                      kpm, epm, attn);
  ln_kernel<<<BB * TT, 256, 0, stream>>>(attn, gamma, beta, lnout);
  gemm_kernel<<<dim3((BB * TT) / 16, CC / 64), 32, 0, stream>>>(lnout, Wt2, out);
}